// CostVolume_15796889715132
// MI455X (gfx1250) — compile-verified
//
#include <hip/hip_runtime.h>

typedef __attribute__((ext_vector_type(2))) float v2f;
typedef __attribute__((ext_vector_type(8))) float v8f;

#define BQ 4
#define GQ 4
#define CGQ 128
#define HQ 128
#define WWQ 256
#define DQ 49
#define KCHUNK 16
#define NCHUNK (CGQ / KCHUNK)   /* 8 */
#define LDS_STRIDE 272          /* 256 + 16 pad: rows 2 apart land on disjoint bank sets */
#define TILES_PER_WAVE 8

__global__ __launch_bounds__(256) void cost_volume_wmma_f32(
    const float* __restrict__ x, const float* __restrict__ y,
    float* __restrict__ out)
{
    __shared__ __align__(16) float Xs[KCHUNK * LDS_STRIDE];
    __shared__ __align__(16) float Ys[KCHUNK * LDS_STRIDE];

    const int tid  = threadIdx.x;
    const int lane = tid & 31;
    const int m    = lane & 15;                         // M (A) / N (B) index
    const int kq   = (lane >> 4) << 1;                  // 0 or 2: K sub-slot per half-wave
    const int wv   = __builtin_amdgcn_readfirstlane(tid >> 5);  // wave id, forced scalar

    const int h  = blockIdx.x & (HQ - 1);
    const int g  = (blockIdx.x >> 7) & (GQ - 1);
    const int bi = (int)(blockIdx.x >> 9);

    const size_t plane = (size_t)HQ * WWQ;              // 32768 floats
    const size_t base  = (size_t)(bi * GQ + g) * CGQ * plane + (size_t)h * WWQ;
    const float* xb = x + base;
    const float* yb = y + base;

    v8f acc[TILES_PER_WAVE] = {};                       // f32 accumulators (64 VGPRs)

    for (int kc = 0; kc < NCHUNK; ++kc) {
        // Stage a 256x16 K-slab of X and Y into LDS, coalesced 128-bit loads.
        // 1024 float4 per array / 256 threads = 4 each.
        #pragma unroll
        for (int r = 0; r < 4; ++r) {
            const int q  = r * 256 + tid;               // float4 index in [0,1024)
            const int kk = q >> 6;                      // 64 float4 per 256-wide row
            const int w4 = (q & 63) << 2;
            const size_t goff = (size_t)(kc * KCHUNK + kk) * plane + w4;
            const float4 xv = *(const float4*)(xb + goff);
            const float4 yv = *(const float4*)(yb + goff);
            *(float4*)&Xs[kk * LDS_STRIDE + w4] = xv;
            *(float4*)&Ys[kk * LDS_STRIDE + w4] = yv;
        }
        __syncthreads();

        // Each wave accumulates its 8 band tiles of X * Y^T over this K-slab.
        #pragma unroll
        for (int t = 0; t < TILES_PER_WAVE; ++t) {
            const int tile = wv * TILES_PER_WAVE + t;
            const int wr = tile >> 2;                   // row tile 0..15
            const int wc = wr - (tile & 3);             // col tile, band offset 0..3
            if (wc < 0) continue;                       // wave-uniform (scalar) skip
            #pragma unroll
            for (int k4 = 0; k4 < 4; ++k4) {
                const int kk = (k4 << 2) + kq;
                v2f a, bm;
                // A 16x4 f32: lanes 0-15 hold {K0,K1}, lanes 16-31 hold {K2,K3}
                a.x  = Xs[kk * LDS_STRIDE + wr * 16 + m];
                a.y  = Xs[(kk + 1) * LDS_STRIDE + wr * 16 + m];
                // B 4x16 f32 (= Y^T tile): same K slotting, N across lanes
                bm.x = Ys[kk * LDS_STRIDE + wc * 16 + m];
                bm.y = Ys[(kk + 1) * LDS_STRIDE + wc * 16 + m];
                acc[t] = __builtin_amdgcn_wmma_f32_16x16x4_f32(
                    false, a, false, bm, (short)0, acc[t], false, false);
            }
        }
        __syncthreads();
    }

    // Epilogue: scatter band elements. C/D layout: VGPR v, lane L ->
    // M = v + 8*(L/16), N = L%16.  out[b,g,k,h,w] with k = w - wq.
    const int halfsel = lane >> 4;
    float* ob = out + (size_t)(bi * GQ + g) * DQ * plane + (size_t)h * WWQ;
    #pragma unroll
    for (int t = 0; t < TILES_PER_WAVE; ++t) {
        const int tile = wv * TILES_PER_WAVE + t;
        const int wr = tile >> 2;
        const int wc = wr - (tile & 3);                 // may be negative: acc == 0 -> writes zeros
        #pragma unroll
        for (int v = 0; v < 8; ++v) {
            const int M   = v + 8 * halfsel;
            const int w   = wr * 16 + M;
            const int wq  = wc * 16 + m;
            const int k   = w - wq;
            if (k >= 0 && k < DQ) {
                ob[(size_t)k * plane + w] = acc[t][v];
            }
        }
    }
}

extern "C" void kernel_launch(void* const* d_in, const int* in_sizes, int n_in,
                              void* d_out, int out_size, void* d_ws, size_t ws_size,
                              hipStream_t stream) {
    const float* x = (const float*)d_in[0];
    const float* y = (const float*)d_in[1];
    float* out = (float*)d_out;
    (void)in_sizes; (void)n_in; (void)out_size; (void)d_ws; (void)ws_size;

    dim3 grid(BQ * GQ * HQ);   // 2048 blocks: one (b,g,h) plane each
    dim3 block(256);           // 8 wave32
    cost_volume_wmma_f32<<<grid, block, 0, stream>>>(x, y, out);
}